// GraphConvolutionalNetwork_44659069944171
// MI455X (gfx1250) — compile-verified
//
#include <hip/hip_runtime.h>
#include <hip/hip_bf16.h>

// GCN layer: agg = segment_sum(feature[src], dst); out = relu(agg @ W + b)
// Stage 1: zero agg        (f32x4 stores)
// Stage 2: edge scatter    (1 wave per edge, global_atomic_add_f32)
// Stage 3: GEMM + epilogue (v_wmma_f32_16x16x4_f32, fp32 end-to-end)

#define FEATS 128

typedef float v2f __attribute__((ext_vector_type(2)));
typedef float v8f __attribute__((ext_vector_type(8)));

__global__ void gcn_zero_kernel(float4* __restrict__ p, int n4) {
    int i = blockIdx.x * blockDim.x + threadIdx.x;
    if (i < n4) p[i] = make_float4(0.f, 0.f, 0.f, 0.f);
}

// One wave32 per edge: lane l moves bytes [16l, 16l+16) of the 512-byte feature row.
__global__ __launch_bounds__(256)
void gcn_scatter_kernel(const float* __restrict__ feat,
                        const int* __restrict__ src,
                        const int* __restrict__ dst,
                        float* __restrict__ agg,
                        int nEdges) {
    int wave = (blockIdx.x * blockDim.x + threadIdx.x) >> 5;
    int lane = threadIdx.x & 31;
    if (wave >= nEdges) return;

    int s = src[wave];
    int d = dst[wave];

    const float4* frow = (const float4*)(feat + (long)s * FEATS);
    float4 v = frow[lane];

    float* arow = agg + (long)d * FEATS + lane * 4;
    atomicAdd(arow + 0, v.x);
    atomicAdd(arow + 1, v.y);
    atomicAdd(arow + 2, v.z);
    atomicAdd(arow + 3, v.w);
}

// One wave32 per 16x16 output tile. K=128 -> 32 chained v_wmma_f32_16x16x4_f32.
// A (16x4 f32, 2 VGPRs): lanes 0-15 hold M=lane, K={k,k+1}; lanes 16-31 hold K={k+2,k+3}.
// B (4x16 f32, 2 VGPRs): row-striped across lanes, mirrored K split.
// C/D (16x16 f32, 8 VGPRs): VGPR r, lanes 0-15 -> M=r, N=lane; lanes 16-31 -> M=r+8.
__global__ __launch_bounds__(256)
void gcn_gemm_wmma_kernel(const float* __restrict__ A,     // [M, 128] agg
                          const float* __restrict__ W,     // [128, 128]
                          const float* __restrict__ bias,  // [128]
                          float* __restrict__ out,         // [M, 128]
                          int M) {
    const int tilesN = FEATS / 16;  // 8
    int waveId = (blockIdx.x * blockDim.x + threadIdx.x) >> 5;
    int lane   = threadIdx.x & 31;
    int tileM  = waveId / tilesN;
    int tileN  = waveId % tilesN;
    if (tileM * 16 >= M) return;    // wave-uniform: EXEC stays all-ones

    int half = lane >> 4;           // 0: lanes 0-15, 1: lanes 16-31
    int l15  = lane & 15;
    int kOff = half * 2;            // K sub-offset within each group of 4

    const float* arow = A + (long)(tileM * 16 + l15) * FEATS;
    const float* wcol = W + tileN * 16 + l15;

    v8f c = {};
#pragma unroll
    for (int kk = 0; kk < FEATS; kk += 4) {
        v2f a, b;
        a.x = arow[kk + kOff + 0];
        a.y = arow[kk + kOff + 1];
        b.x = wcol[(kk + kOff + 0) * FEATS];
        b.y = wcol[(kk + kOff + 1) * FEATS];
        // (neg_a, A, neg_b, B, c_mod, C, reuse_a, reuse_b)
        c = __builtin_amdgcn_wmma_f32_16x16x4_f32(false, a, false, b,
                                                  (short)0, c, false, false);
    }

    // Epilogue: bias + ReLU straight from the C/D VGPR layout.
    int col     = tileN * 16 + l15;
    float bv    = bias[col];
    int rowBase = tileM * 16 + half * 8;
#pragma unroll
    for (int r = 0; r < 8; r++) {
        float v = c[r] + bv;
        v = v > 0.f ? v : 0.f;
        out[(long)(rowBase + r) * FEATS + col] = v;
    }
}

extern "C" void kernel_launch(void* const* d_in, const int* in_sizes, int n_in,
                              void* d_out, int out_size, void* d_ws, size_t ws_size,
                              hipStream_t stream) {
    const float* feat = (const float*)d_in[0];
    const int*   src  = (const int*)d_in[1];
    const int*   dst  = (const int*)d_in[2];
    const float* W    = (const float*)d_in[3];
    const float* bias = (const float*)d_in[4];
    float*       out  = (float*)d_out;

    int nNodes = in_sizes[0] / FEATS;
    int nEdges = in_sizes[1];

    size_t aggBytes = (size_t)nNodes * FEATS * sizeof(float);
    // agg in workspace; in-place via d_out is a racy fallback (the 8 tiles of a row band
    // read all 128 columns that the others overwrite, with no barrier).
    float* agg = (ws_size >= aggBytes) ? (float*)d_ws : out;

    // Stage 1: zero agg
    int n4 = nNodes * FEATS / 4;
    gcn_zero_kernel<<<(n4 + 255) / 256, 256, 0, stream>>>((float4*)agg, n4);

    // Stage 2: scatter-add (8 edges per 256-thread block; 1 wave per edge)
    int blocksS = (nEdges + 7) / 8;
    gcn_scatter_kernel<<<blocksS, 256, 0, stream>>>(feat, src, dst, agg, nEdges);

    // Stage 3: WMMA GEMM + bias + ReLU (8 tiles per block; 1 wave per 16x16 tile)
    int numWaves = (nNodes / 16) * (FEATS / 16);
    int blocksG  = (numWaves + 7) / 8;
    gcn_gemm_wmma_kernel<<<blocksG, 256, 0, stream>>>(agg, W, bias, out, nNodes);
}